// GroupAwareEncoder_4355096839065
// MI455X (gfx1250) — compile-verified
//
#include <hip/hip_runtime.h>
#include <stdint.h>
#include <stddef.h>

// ---------------------------------------------------------------------------
// COO SpMM scatter (LightGCN propagation) for MI455X / gfx1250.
// Memory-bound problem (0.25 flop/byte): no dense MM exists, so the CDNA5
// paths used are async global->LDS staging (ASYNCcnt), no-return
// global_atomic_add_f32 at device scope, and global_prefetch for gathers.
// ---------------------------------------------------------------------------

#define TILE   256          // edges staged per block tile
#define BLOCK  256          // 8 wave32 per block
#define NWAVES (BLOCK / 32)
#define EPW    (TILE / NWAVES)   // 32 edges per wave per tile
#define DDIM   64
#define SLOPE  0.5f

#if __has_builtin(__builtin_amdgcn_global_load_async_to_lds_b32)
#define HAVE_ASYNC 1
#else
#define HAVE_ASYNC 0
#endif

typedef __attribute__((address_space(1))) int gas_int;
typedef __attribute__((address_space(3))) int las_int;

__device__ __forceinline__ void async_ld_b32(const void* g, void* l) {
#if HAVE_ASYNC
  __builtin_amdgcn_global_load_async_to_lds_b32((gas_int*)g, (las_int*)l, 0, 0);
#else
  (void)g; (void)l;
#endif
}

__device__ __forceinline__ void wait_async_le3() {
#if HAVE_ASYNC
#if __has_builtin(__builtin_amdgcn_s_wait_asynccnt)
  __builtin_amdgcn_s_wait_asynccnt(3);
#else
  asm volatile("s_wait_asynccnt 3" ::: "memory");
#endif
#endif
}

__device__ __forceinline__ void wait_async_le0() {
#if HAVE_ASYNC
#if __has_builtin(__builtin_amdgcn_s_wait_asynccnt)
  __builtin_amdgcn_s_wait_asynccnt(0);
#else
  asm volatile("s_wait_asynccnt 0" ::: "memory");
#endif
#endif
}

// No-return f32 atomic add, device scope (RMW performed by L2 atomic units).
__device__ __forceinline__ void gatomic_add(float* p, float v) {
  asm volatile("global_atomic_add_f32 %0, %1, off scope:SCOPE_DEV"
               :: "v"((uint64_t)(uintptr_t)p), "v"(v)
               : "memory");
}

// out[dst[k]*64 + d] += val[k] * maybe_leaky(X[src[k]*64 + d])
__global__ void __launch_bounds__(BLOCK)
spmm_scatter_kernel(const int* __restrict__ src, const int* __restrict__ dst,
                    const float* __restrict__ w, const float* __restrict__ X,
                    float* __restrict__ Y, const int nnz, const int leaky_in) {
#if HAVE_ASYNC
  __shared__ int   sS[2][TILE];
  __shared__ int   sD[2][TILE];
  __shared__ float sW[2][TILE];
#endif
  const int tid  = threadIdx.x;
  const int lane = tid & 31;
  const int wv   = tid >> 5;
  const int ntiles = (nnz + TILE - 1) / TILE;
  const int t0     = blockIdx.x;
  const int tstep  = gridDim.x;

  auto stage = [&](int b, int t) -> bool {
#if HAVE_ASYNC
    const int  e = t * TILE + tid;
    const bool wave_active = (t * TILE + (wv << 5)) < nnz;  // lane0 in range
    if (e < nnz) {                                          // EXEC-masked lanes
      async_ld_b32(src + e, &sS[b][tid]);
      async_ld_b32(dst + e, &sD[b][tid]);
      async_ld_b32(w   + e, &sW[b][tid]);
    }
    return wave_active;  // true iff this wave's ASYNCcnt went up by 3
#else
    (void)b; (void)t;
    return false;
#endif
  };

  int  buf = 0;
  bool cur = (t0 < ntiles) ? stage(0, t0) : false;

  for (int t = t0; t < ntiles; t += tstep) {
    const int  tn  = t + tstep;
    const bool nxt = (tn < ntiles) ? stage(buf ^ 1, tn) : false;
#if HAVE_ASYNC
    if (cur) { if (nxt) wait_async_le3(); else wait_async_le0(); }
#endif
    const int ebase = t * TILE + (wv << 5);
    const int lbase = (wv << 5);

#pragma unroll 4
    for (int i = 0; i < EPW; ++i) {
      const int e = ebase + i;
      if (e >= nnz) break;
      int s, d; float v;
#if HAVE_ASYNC
      s = sS[buf][lbase + i];
      d = sD[buf][lbase + i];
      v = sW[buf][lbase + i];
#else
      s = src[e]; d = dst[e]; v = w[e];
#endif
      // software prefetch of a future gather row (global_prefetch_b8)
      const int ip = i + 8;
      if (ip < EPW && ebase + ip < nnz) {
#if HAVE_ASYNC
        const int sp = sS[buf][lbase + ip];
#else
        const int sp = src[ebase + ip];
#endif
        __builtin_prefetch(X + (size_t)sp * DDIM + (lane << 1), 0, 0);
      }

      float2 x = *(const float2*)(X + (size_t)s * DDIM + (lane << 1));
      if (leaky_in) {
        x.x = x.x > 0.f ? x.x : SLOPE * x.x;
        x.y = x.y > 0.f ? x.y : SLOPE * x.y;
      }
      float* yp = Y + (size_t)d * DDIM + (lane << 1);
      gatomic_add(yp,     v * x.x);
      gatomic_add(yp + 1, v * x.y);
    }
    cur = nxt;
    buf ^= 1;
  }
}

// Zero `cnt[0]*scale` floats (count lives on device; graph-capture safe).
__global__ void zero_dyn_kernel(float4* __restrict__ p,
                                const int* __restrict__ cnt, const int scale) {
  const long long n4 = ((long long)cnt[0] * (long long)scale) >> 2;
  const long long stride = (long long)gridDim.x * blockDim.x;
  float4 z; z.x = z.y = z.z = z.w = 0.f;
  for (long long i = (long long)blockIdx.x * blockDim.x + threadIdx.x; i < n4;
       i += stride)
    p[i] = z;
}

// In-place leaky ReLU over n4 float4s.
__global__ void leaky_kernel(float4* __restrict__ p, const long long n4) {
  const long long stride = (long long)gridDim.x * blockDim.x;
  for (long long i = (long long)blockIdx.x * blockDim.x + threadIdx.x; i < n4;
       i += stride) {
    float4 v = p[i];
    v.x = v.x > 0.f ? v.x : SLOPE * v.x;
    v.y = v.y > 0.f ? v.y : SLOPE * v.y;
    v.z = v.z > 0.f ? v.z : SLOPE * v.z;
    v.w = v.w > 0.f ? v.w : SLOPE * v.w;
    p[i] = v;
  }
}

extern "C" void kernel_launch(void* const* d_in, const int* in_sizes, int n_in,
                              void* d_out, int out_size, void* d_ws,
                              size_t ws_size, hipStream_t stream) {
  const float* embs    = (const float*)d_in[0];  // (N, 64)
  const float* vals    = (const float*)d_in[1];  // (NNZ,)
  const int*   rows    = (const int*)d_in[2];    // (NNZ,) in [0, N)
  const int*   cols    = (const int*)d_in[3];    // (NNZ,) in [0, E)
  const int*   n_nodes = (const int*)d_in[5];    // device scalar N
  const int*   n_edges = (const int*)d_in[6];    // device scalar E
  const int    nnz     = in_sizes[1];

  float* h   = (float*)d_ws;   // (E, 64) intermediate, 25.6 MB
  float* out = (float*)d_out;  // (N, 64) accumulator + final output

  const int ntiles = (nnz + TILE - 1) / TILE;
  int sgrid = ntiles < 2048 ? ntiles : 2048;  // >1 tile/block => pipelining
  if (sgrid < 1) sgrid = 1;
  const int zgrid = 1024;

  // ---- layer 1 ----
  zero_dyn_kernel<<<zgrid, 256, 0, stream>>>((float4*)h, n_edges, DDIM);
  spmm_scatter_kernel<<<sgrid, BLOCK, 0, stream>>>(rows, cols, vals, embs, h,
                                                   nnz, 0);
  zero_dyn_kernel<<<zgrid, 256, 0, stream>>>((float4*)out, n_nodes, DDIM);
  spmm_scatter_kernel<<<sgrid, BLOCK, 0, stream>>>(cols, rows, vals, h, out,
                                                   nnz, 0);
  // ---- layer 2 (leaky of layer-1 output fused into the gather) ----
  zero_dyn_kernel<<<zgrid, 256, 0, stream>>>((float4*)h, n_edges, DDIM);
  spmm_scatter_kernel<<<sgrid, BLOCK, 0, stream>>>(rows, cols, vals, out, h,
                                                   nnz, 1);
  zero_dyn_kernel<<<zgrid, 256, 0, stream>>>((float4*)out, n_nodes, DDIM);
  spmm_scatter_kernel<<<sgrid, BLOCK, 0, stream>>>(cols, rows, vals, h, out,
                                                   nnz, 0);
  // ---- final activation (reference returns leaky(h2) split user/item,
  //      which concatenated is exactly the full (N,64) matrix) ----
  leaky_kernel<<<zgrid, 256, 0, stream>>>((float4*)out, (long long)out_size / 4);
}